// EncodingLayer_56504589746588
// MI455X (gfx1250) — compile-verified
//
#include <hip/hip_runtime.h>
#include <stdint.h>

#ifndef __has_builtin
#define __has_builtin(x) 0
#endif

#if __has_builtin(__builtin_amdgcn_global_load_async_to_lds_b128) && \
    __has_builtin(__builtin_amdgcn_global_load_async_to_lds_b32) &&  \
    __has_builtin(__builtin_amdgcn_s_wait_asynccnt)
#define USE_ASYNC_LDS 1
#else
#define USE_ASYNC_LDS 0
#endif

#define FEAT 8
#define TILE 256
#define INTERP_GRID 1024

// Typed address-space pointers matching the gfx1250 async-to-LDS builtin
// signatures: b128 takes (global int4*, lds int4*, imm off, imm cpol),
// b32 takes (global int*, lds int*, imm off, imm cpol).
typedef int v4i __attribute__((vector_size(16)));
typedef __attribute__((address_space(1))) v4i* g_v4i_p;
typedef __attribute__((address_space(3))) v4i* l_v4i_p;
typedef __attribute__((address_space(1))) int* g_i32_p;
typedef __attribute__((address_space(3))) int* l_i32_p;

__device__ __forceinline__ void async_copy_b128(const void* gsrc, void* ldst) {
#if USE_ASYNC_LDS
  __builtin_amdgcn_global_load_async_to_lds_b128((g_v4i_p)(uintptr_t)gsrc,
                                                 (l_v4i_p)(uintptr_t)ldst, 0, 0);
#else
  (void)gsrc; (void)ldst;
#endif
}

__device__ __forceinline__ void async_copy_b32(const void* gsrc, void* ldst) {
#if USE_ASYNC_LDS
  __builtin_amdgcn_global_load_async_to_lds_b32((g_i32_p)(uintptr_t)gsrc,
                                                (l_i32_p)(uintptr_t)ldst, 0, 0);
#else
  (void)gsrc; (void)ldst;
#endif
}

// ---------------------------------------------------------------------------
// Kernel 1: g[v][0..7] = feats0[map0[v]] + feats1[map1[v]] + feats2[map2[v]] + feats3[map3[v]]
// 200K vertices, collapses 12 gathers/sample later into 3.
// ---------------------------------------------------------------------------
__global__ void build_vertex_table(const float* __restrict__ f0, const int* __restrict__ m0,
                                   const float* __restrict__ f1, const int* __restrict__ m1,
                                   const float* __restrict__ f2, const int* __restrict__ m2,
                                   const float* __restrict__ f3, const int* __restrict__ m3,
                                   float* __restrict__ gtab, int nv) {
  const int v = blockIdx.x * blockDim.x + threadIdx.x;
  if (v >= nv) return;
  const float4* p0 = (const float4*)(f0 + (size_t)m0[v] * FEAT);
  const float4* p1 = (const float4*)(f1 + (size_t)m1[v] * FEAT);
  const float4* p2 = (const float4*)(f2 + (size_t)m2[v] * FEAT);
  const float4* p3 = (const float4*)(f3 + (size_t)m3[v] * FEAT);
  float4 x0 = p0[0], y0 = p0[1];
  float4 x1 = p1[0], y1 = p1[1];
  float4 x2 = p2[0], y2 = p2[1];
  float4 x3 = p3[0], y3 = p3[1];
  float4 xs, ys;
  xs.x = ((x0.x + x1.x) + x2.x) + x3.x;
  xs.y = ((x0.y + x1.y) + x2.y) + x3.y;
  xs.z = ((x0.z + x1.z) + x2.z) + x3.z;
  xs.w = ((x0.w + x1.w) + x2.w) + x3.w;
  ys.x = ((y0.x + y1.x) + y2.x) + y3.x;
  ys.y = ((y0.y + y1.y) + y2.y) + y3.y;
  ys.z = ((y0.z + y1.z) + y2.z) + y3.z;
  ys.w = ((y0.w + y1.w) + y2.w) + y3.w;
  float4* op = (float4*)(gtab + (size_t)v * FEAT);
  op[0] = xs;
  op[1] = ys;
}

// ---------------------------------------------------------------------------
// Stage one tile of bary (count*3 floats) and triangle (count*3 ints) into LDS
// via CDNA5 async global->LDS copies (ASYNCcnt-tracked), b128 granularity.
// ---------------------------------------------------------------------------
__device__ __forceinline__ void stage_tile(const float* __restrict__ bary,
                                           const int* __restrict__ tri,
                                           float* sb, int* st,
                                           long s0, int count, int tid) {
  const int ndw = count * 3;
  const int nb = ndw >> 2;  // number of 16B chunks
  const float* gb = bary + s0 * 3;
  const int*   gt = tri  + s0 * 3;
#if USE_ASYNC_LDS
  for (int i = tid; i < nb; i += TILE) {
    async_copy_b128(gb + i * 4, sb + i * 4);
    async_copy_b128(gt + i * 4, st + i * 4);
  }
  for (int i = (nb << 2) + tid; i < ndw; i += TILE) {
    async_copy_b32(gb + i, sb + i);
    async_copy_b32(gt + i, st + i);
  }
#else
  for (int i = tid; i < ndw; i += TILE) {
    sb[i] = gb[i];
    st[i] = gt[i];
  }
#endif
}

// ---------------------------------------------------------------------------
// Kernel 2: out[n,:] = sum_c bary[n,c] * g[triangle[n,c], :]
// One thread per sample; streaming inputs double-buffered through LDS with
// async copies; random 32B gathers from the 6.4MB (L2-resident) vertex table.
// ---------------------------------------------------------------------------
__global__ void interp_kernel(const float* __restrict__ bary, const int* __restrict__ tri,
                              const float* __restrict__ gtab, float* __restrict__ out, int n) {
  __shared__ __align__(16) float sb[2][TILE * 3];
  __shared__ __align__(16) int   st[2][TILE * 3];
  const int tid = threadIdx.x;
  const int nTiles = (n + TILE - 1) / TILE;
  int t = blockIdx.x;
  if (t >= nTiles) return;

  int cnt0 = n - t * TILE;
  if (cnt0 > TILE) cnt0 = TILE;
  stage_tile(bary, tri, sb[0], st[0], (long)t * TILE, cnt0, tid);

  int buf = 0;
  const int stride = gridDim.x;
  for (; t < nTiles; t += stride) {
#if USE_ASYNC_LDS
    __builtin_amdgcn_s_wait_asynccnt(0);
#endif
    __syncthreads();  // current buffer fully staged & visible block-wide

    const int nt = t + stride;
    if (nt < nTiles) {
      int c2 = n - nt * TILE;
      if (c2 > TILE) c2 = TILE;
      stage_tile(bary, tri, sb[buf ^ 1], st[buf ^ 1], (long)nt * TILE, c2, tid);
    }

    const long s = (long)t * TILE + tid;
    if (s < n) {
      float a0 = 0.f, a1 = 0.f, a2 = 0.f, a3 = 0.f;
      float a4 = 0.f, a5 = 0.f, a6 = 0.f, a7 = 0.f;
#pragma unroll
      for (int c = 0; c < 3; ++c) {
        const float w = sb[buf][tid * 3 + c];
        const int   v = st[buf][tid * 3 + c];
        const float4* gp = (const float4*)(gtab + (size_t)v * FEAT);
        const float4 x = gp[0];
        const float4 y = gp[1];
        a0 = fmaf(w, x.x, a0); a1 = fmaf(w, x.y, a1);
        a2 = fmaf(w, x.z, a2); a3 = fmaf(w, x.w, a3);
        a4 = fmaf(w, y.x, a4); a5 = fmaf(w, y.y, a5);
        a6 = fmaf(w, y.z, a6); a7 = fmaf(w, y.w, a7);
      }
      float4* op = (float4*)(out + s * FEAT);
      op[0] = make_float4(a0, a1, a2, a3);
      op[1] = make_float4(a4, a5, a6, a7);
    }
    __syncthreads();  // all readers done before this buffer is overwritten
    buf ^= 1;
  }
}

// ---------------------------------------------------------------------------
// Fallback: fully fused (used only if ws_size can't hold the vertex table).
// ---------------------------------------------------------------------------
__global__ void fused_kernel(const float* __restrict__ bary, const int* __restrict__ tri,
                             const float* __restrict__ f0, const int* __restrict__ m0,
                             const float* __restrict__ f1, const int* __restrict__ m1,
                             const float* __restrict__ f2, const int* __restrict__ m2,
                             const float* __restrict__ f3, const int* __restrict__ m3,
                             float* __restrict__ out, int n) {
  const long s = (long)blockIdx.x * blockDim.x + threadIdx.x;
  if (s >= n) return;
  float a0 = 0.f, a1 = 0.f, a2 = 0.f, a3 = 0.f;
  float a4 = 0.f, a5 = 0.f, a6 = 0.f, a7 = 0.f;
#pragma unroll
  for (int c = 0; c < 3; ++c) {
    const float w = bary[s * 3 + c];
    const int   v = tri[s * 3 + c];
    const float4* p0 = (const float4*)(f0 + (size_t)m0[v] * FEAT);
    const float4* p1 = (const float4*)(f1 + (size_t)m1[v] * FEAT);
    const float4* p2 = (const float4*)(f2 + (size_t)m2[v] * FEAT);
    const float4* p3 = (const float4*)(f3 + (size_t)m3[v] * FEAT);
    float4 x0 = p0[0], y0 = p0[1];
    float4 x1 = p1[0], y1 = p1[1];
    float4 x2 = p2[0], y2 = p2[1];
    float4 x3 = p3[0], y3 = p3[1];
    const float gx0 = ((x0.x + x1.x) + x2.x) + x3.x;
    const float gx1 = ((x0.y + x1.y) + x2.y) + x3.y;
    const float gx2 = ((x0.z + x1.z) + x2.z) + x3.z;
    const float gx3 = ((x0.w + x1.w) + x2.w) + x3.w;
    const float gy0 = ((y0.x + y1.x) + y2.x) + y3.x;
    const float gy1 = ((y0.y + y1.y) + y2.y) + y3.y;
    const float gy2 = ((y0.z + y1.z) + y2.z) + y3.z;
    const float gy3 = ((y0.w + y1.w) + y2.w) + y3.w;
    a0 = fmaf(w, gx0, a0); a1 = fmaf(w, gx1, a1);
    a2 = fmaf(w, gx2, a2); a3 = fmaf(w, gx3, a3);
    a4 = fmaf(w, gy0, a4); a5 = fmaf(w, gy1, a5);
    a6 = fmaf(w, gy2, a6); a7 = fmaf(w, gy3, a7);
  }
  float4* op = (float4*)(out + s * FEAT);
  op[0] = make_float4(a0, a1, a2, a3);
  op[1] = make_float4(a4, a5, a6, a7);
}

extern "C" void kernel_launch(void* const* d_in, const int* in_sizes, int n_in,
                              void* d_out, int out_size, void* d_ws, size_t ws_size,
                              hipStream_t stream) {
  // setup_inputs order: bary, triangle, feats0, map0, feats1, map1, feats2, map2, feats3, map3
  const float* bary = (const float*)d_in[0];
  const int*   tri  = (const int*)d_in[1];
  const float* f0 = (const float*)d_in[2];
  const int*   m0 = (const int*)d_in[3];
  const float* f1 = (const float*)d_in[4];
  const int*   m1 = (const int*)d_in[5];
  const float* f2 = (const float*)d_in[6];
  const int*   m2 = (const int*)d_in[7];
  const float* f3 = (const float*)d_in[8];
  const int*   m3 = (const int*)d_in[9];
  float* out = (float*)d_out;

  const int n  = in_sizes[0] / 3;   // 2,000,000 samples
  const int nv = in_sizes[3];       // 200,000 finest vertices (len of map0)

  const size_t need = (size_t)nv * FEAT * sizeof(float);
  if (ws_size >= need) {
    float* gtab = (float*)d_ws;
    build_vertex_table<<<(nv + 255) / 256, 256, 0, stream>>>(
        f0, m0, f1, m1, f2, m2, f3, m3, gtab, nv);
    const int nTiles = (n + TILE - 1) / TILE;
    const int grid = nTiles < INTERP_GRID ? nTiles : INTERP_GRID;
    interp_kernel<<<grid, TILE, 0, stream>>>(bary, tri, gtab, out, n);
  } else {
    fused_kernel<<<(n + 255) / 256, 256, 0, stream>>>(
        bary, tri, f0, m0, f1, m1, f2, m2, f3, m3, out, n);
  }
}